// ContextOnlySoftDotAttention_20667382628536
// MI455X (gfx1250) — compile-verified
//
#include <hip/hip_runtime.h>
#include <math.h>

#define B_    256
#define S_    2048
#define DIM_  1024
#define CDIM_ 512

typedef __attribute__((ext_vector_type(2))) float v2f;
typedef __attribute__((ext_vector_type(8))) float v8f;

// ---------------------------------------------------------------------------
// Kernel 1: target[B, CDIM] = h[B, DIM] @ W[CDIM, DIM]^T via V_WMMA_F32_16X16X4_F32
// One wave per 16x16 output tile. 16 M-tiles x 32 N-tiles = 512 tiles.
// A (16x4, f32): lanes 0-15 -> M=lane, VGPR0=K0, VGPR1=K1; lanes 16-31 -> K2/K3.
// B (4x16, f32): lanes 0-15 -> N=lane, K0/K1; lanes 16-31 -> K2/K3.
// C/D (16x16, f32): VGPR r -> M = r + 8*(lane>=16), N = lane&15.
// ---------------------------------------------------------------------------
__global__ __launch_bounds__(256) void linear_wmma(const float* __restrict__ h,
                                                   const float* __restrict__ W,
                                                   float* __restrict__ target) {
  const int lane = threadIdx.x & 31;
  const int wave = threadIdx.x >> 5;
  const int tile = blockIdx.x * 8 + wave;   // 0..511
  const int tm = tile >> 5;                 // 0..15
  const int tn = tile & 31;                 // 0..31
  const int m0 = tm * 16, n0 = tn * 16;

  const int half = lane >> 4;               // 0 or 1
  const int l16  = lane & 15;
  const float* __restrict__ aRow = h + (size_t)(m0 + l16) * DIM_ + half * 2;
  const float* __restrict__ bRow = W + (size_t)(n0 + l16) * DIM_ + half * 2;

  v8f acc = {};
#pragma unroll 4
  for (int k0 = 0; k0 < DIM_; k0 += 4) {
    v2f a = *(const v2f*)(aRow + k0);
    v2f b = *(const v2f*)(bRow + k0);
    acc = __builtin_amdgcn_wmma_f32_16x16x4_f32(false, a, false, b,
                                                (short)0, acc, false, false);
  }

#pragma unroll
  for (int r = 0; r < 8; ++r) {
    const int row = m0 + r + 8 * half;
    target[(size_t)row * CDIM_ + n0 + l16] = acc[r];
  }
}

// ---------------------------------------------------------------------------
// Kernel 2: fused attention. One 512-thread block (16 wave32) per batch b.
//   P1: attn_raw[s] = context[b,s,:] . target[b,:]  (wave per row, coalesced)
//   P2: masked softmax over S in LDS, write attn out
//   P3: weighted_context[c] = sum_s attn[s]*context[b,s,c]  (L2-resident reread)
// ---------------------------------------------------------------------------
__global__ __launch_bounds__(512) void attn_fused(const float* __restrict__ ctx,
                                                  const unsigned char* __restrict__ mask,
                                                  const float* __restrict__ target,
                                                  float* __restrict__ out_wc,
                                                  float* __restrict__ out_attn) {
  __shared__ float  sT[CDIM_];        // target row (2 KB)
  __shared__ float  sA[S_];           // scores / probs (8 KB)
  __shared__ float  sRed[16];         // per-wave reduce slots
  __shared__ float4 sAcc[4][CDIM_/4]; // phase-3 partials (8 KB)

  const int b    = blockIdx.x;
  const int tid  = threadIdx.x;
  const int lane = tid & 31;
  const int wave = tid >> 5;          // 0..15
  const float* __restrict__ cb = ctx + (size_t)b * S_ * CDIM_;

  sT[tid] = target[(size_t)b * CDIM_ + tid];
  __syncthreads();

  // ---- Phase 1: raw scores, wave per row ----
  for (int i = 0; i < S_ / 16; ++i) {
    const int s = wave + 16 * i;
    const float* __restrict__ row = cb + (size_t)s * CDIM_;
    __builtin_prefetch(row + 16 * CDIM_, 0, 1);   // next row chunk for this wave
    float p = 0.f;
#pragma unroll
    for (int q = 0; q < 4; ++q) {
      const int c4 = lane + 32 * q;
      const float4 v = ((const float4*)row)[c4];
      const float4 t = ((const float4*)sT)[c4];
      p += v.x * t.x + v.y * t.y + v.z * t.z + v.w * t.w;
    }
#pragma unroll
    for (int off = 16; off > 0; off >>= 1)
      p += __shfl_xor(p, off, 32);
    if (lane == 0)
      sA[s] = mask[(size_t)b * S_ + s] ? -INFINITY : p;
  }
  __syncthreads();

  // ---- Phase 2: softmax over S ----
  float m = -INFINITY;
#pragma unroll
  for (int j = 0; j < S_ / 512; ++j) m = fmaxf(m, sA[tid + 512 * j]);
#pragma unroll
  for (int off = 16; off > 0; off >>= 1) m = fmaxf(m, __shfl_xor(m, off, 32));
  if (lane == 0) sRed[wave] = m;
  __syncthreads();
  if (tid == 0) {
    float mm = sRed[0];
#pragma unroll
    for (int w = 1; w < 16; ++w) mm = fmaxf(mm, sRed[w]);
    sRed[0] = mm;
  }
  __syncthreads();
  m = sRed[0];
  __syncthreads();

  float sum = 0.f;
#pragma unroll
  for (int j = 0; j < S_ / 512; ++j) {
    const int i = tid + 512 * j;
    const float e = __expf(sA[i] - m);
    sA[i] = e;
    sum += e;
  }
#pragma unroll
  for (int off = 16; off > 0; off >>= 1) sum += __shfl_xor(sum, off, 32);
  if (lane == 0) sRed[wave] = sum;
  __syncthreads();
  if (tid == 0) {
    float ss = 0.f;
#pragma unroll
    for (int w = 0; w < 16; ++w) ss += sRed[w];
    sRed[0] = 1.0f / ss;
  }
  __syncthreads();
  const float inv = sRed[0];
#pragma unroll
  for (int j = 0; j < S_ / 512; ++j) {
    const int i = tid + 512 * j;
    const float a = sA[i] * inv;
    sA[i] = a;
    out_attn[(size_t)b * S_ + i] = a;
  }
  __syncthreads();

  // ---- Phase 3: weighted context (context re-read hits L2) ----
  const int g = tid >> 7;    // 0..3 : s-range group
  const int q = tid & 127;   // column quad -> columns 4q..4q+3
  float4 acc = {0.f, 0.f, 0.f, 0.f};
  const int sBeg = g * (S_ / 4);
  const int sEnd = sBeg + (S_ / 4);
  for (int s = sBeg; s < sEnd; ++s) {
    const float w = sA[s];
    const float* __restrict__ row = cb + (size_t)s * CDIM_;
    __builtin_prefetch(row + 8 * CDIM_, 0, 1);
    const float4 v = ((const float4*)row)[q];
    acc.x += w * v.x; acc.y += w * v.y; acc.z += w * v.z; acc.w += w * v.w;
  }
  sAcc[g][q] = acc;
  __syncthreads();
  if (g == 0) {
    const float4 a0 = sAcc[0][q], a1 = sAcc[1][q], a2 = sAcc[2][q], a3 = sAcc[3][q];
    float4 r;
    r.x = a0.x + a1.x + a2.x + a3.x;
    r.y = a0.y + a1.y + a2.y + a3.y;
    r.z = a0.z + a1.z + a2.z + a3.z;
    r.w = a0.w + a1.w + a2.w + a3.w;
    ((float4*)(out_wc + (size_t)b * CDIM_))[q] = r;
  }
}

extern "C" void kernel_launch(void* const* d_in, const int* in_sizes, int n_in,
                              void* d_out, int out_size, void* d_ws, size_t ws_size,
                              hipStream_t stream) {
  const float* h            = (const float*)d_in[0];   // [B, DIM]
  const float* ctx          = (const float*)d_in[1];   // [B, S, CDIM]
  const unsigned char* mask = (const unsigned char*)d_in[2]; // [B, S] bool
  const float* W            = (const float*)d_in[3];   // [CDIM, DIM]

  float* target   = (float*)d_ws;                // [B, CDIM] scratch
  float* out_wc   = (float*)d_out;               // [B, CDIM]
  float* out_attn = out_wc + (size_t)B_ * CDIM_; // [B, S]

  linear_wmma<<<64, 256, 0, stream>>>(h, W, target);
  attn_fused<<<B_, 512, 0, stream>>>(ctx, mask, target, out_wc, out_attn);
}